// E3LinearChannelMixing_29609504538702
// MI455X (gfx1250) — compile-verified
//
#include <hip/hip_runtime.h>
#include <stdint.h>

// E3 linear channel mixing on gfx1250 (MI455X):
// 120 batched GEMMs  Y[b,o,c] = sum_i x[b,i,c] * W_{g(c)}[m(c)][o,i]  (+bias c<32)
// V_WMMA_F32_16X16X4_F32 + double-buffered GLOBAL_LOAD_ASYNC_TO_LDS_B128 staging,
// non-temporal packed B96 output stores (y is write-once; keep x/W in L2).

typedef float v2f __attribute__((ext_vector_type(2)));
typedef float v3f __attribute__((ext_vector_type(3)));
typedef float v8f __attribute__((ext_vector_type(8)));

namespace {
constexpr int kNIn  = 256;
constexpr int kNOut = 256;
constexpr int kC    = 120;   // 32*1 + 16*3 + 8*5
constexpr int kB    = 1024;

constexpr int BM = 32;   // batch rows per block
constexpr int BN = 32;   // output cols per block
constexpr int CB = 24;   // irrep channels per block
constexpr int KI = 8;    // K staged per chunk

constexpr int XROW = KI * CB + 4;   // 196 dwords: x row = 784 B, 16B-aligned
constexpr int WROW = KI + 4;        // 12 dwords: weight o-row = 48 B, 16B-aligned
constexpr int WCH  = BN * WROW;     // 384 dwords per channel plane
constexpr int XBUF = BM * XROW;     // 6272 dwords per buffer
constexpr int WBUF = CB * WCH;      // 9216 dwords per buffer
constexpr int NTHREADS = 256;       // 8 wave32 waves
constexpr int NCHUNK = kNIn / KI;   // 32 K-chunks (even)
}  // namespace

// Async DMA global->LDS, 16B per lane. GVS form: scalar 64b base + per-lane
// 32b offset. GV form: per-lane 64b address (needed for mixed W0/W1/W2 rows).
#define ASYNC_X(ldsv, offv, basev, IMM)                                      \
  asm volatile("global_load_async_to_lds_b128 %0, %1, %2 offset:" #IMM       \
               : : "v"(ldsv), "v"(offv), "s"(basev) : "memory")
#define ASYNC_W(ldsv, addrv, IMM)                                            \
  asm volatile("global_load_async_to_lds_b128 %0, %1, off offset:" #IMM      \
               : : "v"(ldsv), "v"(addrv) : "memory")
#define WAIT_ASYNC(IMM) \
  asm volatile("s_wait_asynccnt " #IMM : : : "memory")

__global__ __launch_bounds__(NTHREADS)
void e3linear_wmma_async_kernel(const float* __restrict__ x,   // [B, NIn, 120]
                                const float* __restrict__ W0,  // [32, NOut, NIn]
                                const float* __restrict__ W1,  // [16, NOut, NIn]
                                const float* __restrict__ W2,  // [ 8, NOut, NIn]
                                const float* __restrict__ b0,  // [32, NOut]
                                float* __restrict__ out) {     // [B, NOut, 120]
  __shared__ float xs[2][XBUF];
  __shared__ float wsb[2][WBUF];

  const int btile = blockIdx.x * BM;  // gridDim.x = 32
  const int otile = blockIdx.y * BN;  // gridDim.y = 8
  const int ctile = blockIdx.z * CB;  // gridDim.z = 5

  const int tid  = threadIdx.x;
  const int wave = tid >> 5;
  const int lane = tid & 31;
  const int lrow  = lane & 15;
  const int khalf = (lane >> 4) * 2;  // lanes 16..31 hold K+2, K+3

  // ---- One-time staging assignment (shifts/masks only in the hot loop). ----
  // x: thread t owns row (bb = t>>3, ki = t&7): 24 floats = 6 x 16B segments.
  const int bb = tid >> 3;
  const int ki = tid & 7;
  uint32_t xoff = (uint32_t)(((btile + bb) * (kNIn * kC) + ki * kC + ctile) * 4);
  const uint64_t xbase = (uint64_t)(uintptr_t)x;
  const uint32_t xlds0 = (uint32_t)(uintptr_t)(&xs[0][bb * XROW + ki * CB]);

  // W: thread t owns 3 rows (cc = r>>5, oo = r&31): 8 floats = 2 x 16B segments.
  uint64_t waddr[3];
  uint32_t wlds0[3];
#pragma unroll
  for (int j = 0; j < 3; ++j) {
    const int r  = tid + j * NTHREADS;
    const int cc = r >> 5;
    const int oo = r & 31;
    const int c  = ctile + cc;
    int m;
    const float* Wg;
    if (c < 32)      { m = c;            Wg = W0; }
    else if (c < 80) { m = (c - 32) / 3; Wg = W1; }
    else             { m = (c - 80) / 5; Wg = W2; }
    waddr[j] = (uint64_t)(uintptr_t)(Wg + ((size_t)m * kNOut + (otile + oo)) * kNIn);
    wlds0[j] = (uint32_t)(uintptr_t)(&wsb[0][cc * WCH + oo * WROW]);
  }

  // Accumulators: wave owns channels [3*wave, 3*wave+3), 2x2 16x16 subtiles.
  v8f acc[3][2][2];
#pragma unroll
  for (int a = 0; a < 3; ++a)
#pragma unroll
    for (int b = 0; b < 2; ++b)
#pragma unroll
      for (int o = 0; o < 2; ++o)
#pragma unroll
        for (int r = 0; r < 8; ++r) acc[a][b][o][r] = 0.0f;

  // Issue one K-chunk's DMA: 12 async B128 per wave (6 x-tile + 6 weights).
  // BUF is a compile-time literal at every call site -> LDS addrs fold.
  auto issue = [&](const int BUF) {
    const uint32_t xl = xlds0 + (uint32_t)(BUF * XBUF * 4);
    ASYNC_X(xl, xoff, xbase, 0);
    ASYNC_X(xl, xoff, xbase, 16);
    ASYNC_X(xl, xoff, xbase, 32);
    ASYNC_X(xl, xoff, xbase, 48);
    ASYNC_X(xl, xoff, xbase, 64);
    ASYNC_X(xl, xoff, xbase, 80);
    xoff += (uint32_t)(KI * kC * 4);  // advance 8 input channels = 3840 B
#pragma unroll
    for (int j = 0; j < 3; ++j) {
      const uint32_t wl = wlds0[j] + (uint32_t)(BUF * WBUF * 4);
      ASYNC_W(wl, waddr[j], 0);
      ASYNC_W(wl, waddr[j], 16);
      waddr[j] += (uint64_t)(KI * 4);  // advance 8 K = 32 B
    }
  };

  // Compute one K-chunk: 3 channels x 2 k-steps x 2x2 subtiles = 24 WMMA.
  auto compute = [&](const int BUF) {
    const float* xb = &xs[BUF][0];
    const float* wb = &wsb[BUF][0];
#pragma unroll
    for (int cl = 0; cl < 3; ++cl) {
      const int cc = wave * 3 + cl;
      const float* xcol = &xb[lrow * XROW + cc];
      const float* wcol = &wb[cc * WCH + lrow * WROW];
#pragma unroll
      for (int kk = 0; kk < KI; kk += 4) {
        const int kb = kk + khalf;
        v2f afr[2], bfr[2];
#pragma unroll
        for (int os = 0; os < 2; ++os) {  // B 4x16: lane = N col, 2 K vals
          bfr[os].x = wcol[os * 16 * WROW + kb];
          bfr[os].y = wcol[os * 16 * WROW + kb + 1];
        }
#pragma unroll
        for (int bs = 0; bs < 2; ++bs) {  // A 16x4: lane = M row, 2 K vals
          afr[bs].x = xcol[bs * 16 * XROW + kb * CB];
          afr[bs].y = xcol[bs * 16 * XROW + kb * CB + CB];
        }
#pragma unroll
        for (int bs = 0; bs < 2; ++bs)
#pragma unroll
          for (int os = 0; os < 2; ++os)
            acc[cl][bs][os] = __builtin_amdgcn_wmma_f32_16x16x4_f32(
                false, afr[bs], false, bfr[os], (short)0, acc[cl][bs][os],
                false, false);
      }
    }
  };

  // ---- Double-buffered pipeline, explicitly unrolled x2 so the buffer index
  //      is a literal everywhere (no dynamic LDS base re-derivation). ----
  issue(0);  // chunk 0 -> buf0
#pragma unroll 1
  for (int n = 0; n < NCHUNK - 2; n += 2) {
    issue(1);        // chunk n+1 -> buf1
    WAIT_ASYNC(12);  // chunk n resident (async loads retire in order)
    __syncthreads();
    compute(0);      // chunk n
    __syncthreads();
    issue(0);        // chunk n+2 -> buf0 (safe: all waves done reading buf0)
    WAIT_ASYNC(12);  // chunk n+1 resident
    __syncthreads();
    compute(1);      // chunk n+1
    __syncthreads();
  }
  issue(1);          // chunk NCHUNK-1 -> buf1
  WAIT_ASYNC(12);
  __syncthreads();
  compute(0);        // chunk NCHUNK-2
  __syncthreads();
  WAIT_ASYNC(0);
  __syncthreads();
  compute(1);        // chunk NCHUNK-1

  // ---- Epilogue: fused bias; the wave's 3 channels are contiguous in memory,
  //      so pack them into one non-temporal B96 store per (b,o,r). ----
  const int cbase = ctile + wave * 3;
#pragma unroll
  for (int os = 0; os < 2; ++os) {
    const int o = otile + os * 16 + lrow;
    float bias[3];
#pragma unroll
    for (int cl = 0; cl < 3; ++cl) {
      const int c = cbase + cl;
      bias[cl] = (c < 32) ? b0[c * kNOut + o] : 0.0f;  // may straddle c==32
    }
#pragma unroll
    for (int bs = 0; bs < 2; ++bs) {
#pragma unroll
      for (int r = 0; r < 8; ++r) {
        // C/D layout: VGPR r -> M=r (lanes 0-15) / M=r+8 (lanes 16-31)
        const int mrow = (lane < 16) ? r : r + 8;
        const int bbq = btile + bs * 16 + mrow;
        v3f v;
        v.x = acc[0][bs][os][r] + bias[0];
        v.y = acc[1][bs][os][r] + bias[1];
        v.z = acc[2][bs][os][r] + bias[2];
        float* p = out + (size_t)bbq * (kNOut * kC) + (size_t)o * kC + cbase;
        __builtin_nontemporal_store(v, (v3f*)p);  // write-once stream: th=NT
      }
    }
  }
}

extern "C" void kernel_launch(void* const* d_in, const int* in_sizes, int n_in,
                              void* d_out, int out_size, void* d_ws,
                              size_t ws_size, hipStream_t stream) {
  (void)in_sizes; (void)n_in; (void)out_size; (void)d_ws; (void)ws_size;
  const float* x  = (const float*)d_in[0];
  const float* W0 = (const float*)d_in[1];
  const float* W1 = (const float*)d_in[2];
  const float* W2 = (const float*)d_in[3];
  const float* b0 = (const float*)d_in[4];
  float* out = (float*)d_out;

  static_assert(kB % BM == 0 && kNOut % BN == 0 && kC % CB == 0, "tiling");
  static_assert((NCHUNK & 1) == 0 && NCHUNK >= 4, "pipeline");
  dim3 grid(kB / BM, kNOut / BN, kC / CB);  // 32 x 8 x 5 = 1280 blocks
  dim3 block(NTHREADS);
  e3linear_wmma_async_kernel<<<grid, block, 0, stream>>>(x, W0, W1, W2, b0, out);
}